// VariationalGCNEncoder_33432025432489
// MI455X (gfx1250) — compile-verified
//
#include <hip/hip_runtime.h>

typedef __attribute__((ext_vector_type(2))) float v2f;
typedef __attribute__((ext_vector_type(8))) float v8f;
typedef __attribute__((ext_vector_type(4))) int   v4i;

#if __has_builtin(__builtin_amdgcn_global_load_async_to_lds_b128)
#define HAVE_ASYNC_LDS 1
#endif

static __device__ __forceinline__ void wait_async_lds() {
#if __has_builtin(__builtin_amdgcn_s_wait_asynccnt)
  __builtin_amdgcn_s_wait_asynccnt(0);
#else
  asm volatile("s_wait_asynccnt 0x0" ::: "memory");
#endif
}

// ---------------- small utility kernels ----------------

__global__ __launch_bounds__(256) void k_fill(float* __restrict__ p, int n, float v) {
  int i = blockIdx.x * 256 + threadIdx.x;
  if (i < n) p[i] = v;
}

// deg[dst] += 1 for every edge (deg pre-initialized to 1.0 = self loop)
__global__ __launch_bounds__(256) void k_degree(float* __restrict__ deg,
                                                const long long* __restrict__ dst, int E) {
  int i = blockIdx.x * 256 + threadIdx.x;
  if (i < E) atomicAdd(&deg[(int)dst[i]], 1.0f);
}

__global__ __launch_bounds__(256) void k_rsqrt(float* __restrict__ dinv,
                                               const float* __restrict__ deg, int n) {
  int i = blockIdx.x * 256 + threadIdx.x;
  if (i < n) {
    float d = deg[i];
    dinv[i] = (d > 0.0f) ? rsqrtf(d) : 0.0f;
  }
}

// W_cat[128][128] = [W_mu (128x64) | W_lv (128x64)]
__global__ __launch_bounds__(256) void k_wcat(float* __restrict__ Wcat,
                                              const float* __restrict__ Wmu,
                                              const float* __restrict__ Wlv) {
  int i = blockIdx.x * 256 + threadIdx.x;  // 16384 elements
  int r = i >> 7, c = i & 127;
  Wcat[i] = (c < 64) ? Wmu[r * 64 + c] : Wlv[r * 64 + (c - 64)];
}

// ---------------- WMMA GEMM: out[N x 128] = A[N x 128] @ W[128 x 128] ----------------
// Exact fp32 math via V_WMMA_F32_16X16X4_F32.
// Block = 8 wave32s; block covers 64 rows x 128 cols; wave w owns a 16-col slice and
// 4 row-tiles. B fragments are register-resident (loaded once per block) and reused
// across 4 independent WMMA accumulation chains per k-step (keeps the XDL pipe full).
#define WMMA_F32(acc, a, b)                                                     \
  acc = __builtin_amdgcn_wmma_f32_16x16x4_f32(false, (a), false, (b), (short)0, \
                                              (acc), false, false)

__global__ __launch_bounds__(256) void k_gemm128(const float* __restrict__ A,
                                                 const float* __restrict__ W,
                                                 float* __restrict__ out, int N) {
  // 132-float row stride: 16B-aligned rows for B128 async writes (528 % 16 == 0)
  // and conflict-free column reads (banks 4*lane over 16 lanes are distinct).
  __shared__ float xs[64][132];
  const int row0 = blockIdx.x * 64;
  const int tid = threadIdx.x;

  // Stage 64x128 tile of A into LDS: 2048 float4 chunks, 8 per thread.
  for (int ch = tid; ch < 64 * 32; ch += 256) {
    int r = ch >> 5, q = ch & 31;
    int gr = row0 + r;
    if (gr < N) {
#ifdef HAVE_ASYNC_LDS
      __builtin_amdgcn_global_load_async_to_lds_b128(
          (__attribute__((address_space(1))) v4i*)(A + (size_t)gr * 128 + q * 4),
          (__attribute__((address_space(3))) v4i*)&xs[r][q * 4], 0, 0);
#else
      float4 v = ((const float4*)A)[(size_t)gr * 32 + q];
      *(float4*)&xs[r][q * 4] = v;
#endif
    } else {
      float4 z = {0.0f, 0.0f, 0.0f, 0.0f};
      *(float4*)&xs[r][q * 4] = z;
    }
  }
#ifdef HAVE_ASYNC_LDS
  wait_async_lds();  // ASYNCcnt is not covered by the workgroup barrier
#endif
  __syncthreads();

  const int wave = tid >> 5;  // 0..7 -> 16-column slice
  const int lane = tid & 31;
  const int col0 = wave * 16;
  const int mn = lane & 15;   // M index for A, N index for B/C/D
  const int hi = lane >> 4;   // half-wave selector: K pair {0,1} vs {2,3}

  // Preload all B fragments for this wave's column slice (128x16 of W) into regs.
  v2f bf[32];
  #pragma unroll
  for (int kk = 0; kk < 32; ++kk) {
    const int kb = kk * 4 + 2 * hi;
    bf[kk].x = W[(size_t)(kb + 0) * 128 + col0 + mn];
    bf[kk].y = W[(size_t)(kb + 1) * 128 + col0 + mn];
  }

  v8f acc0 = {}, acc1 = {}, acc2 = {}, acc3 = {};
  #pragma unroll
  for (int kk = 0; kk < 32; ++kk) {
    const int kb = kk * 4 + 2 * hi;
    v2f a0, a1, a2, a3;
    a0.x = xs[mn +  0][kb]; a0.y = xs[mn +  0][kb + 1];
    a1.x = xs[mn + 16][kb]; a1.y = xs[mn + 16][kb + 1];
    a2.x = xs[mn + 32][kb]; a2.y = xs[mn + 32][kb + 1];
    a3.x = xs[mn + 48][kb]; a3.y = xs[mn + 48][kb + 1];
    WMMA_F32(acc0, a0, bf[kk]);  // 4 independent chains -> back-to-back WMMAs
    WMMA_F32(acc1, a1, bf[kk]);
    WMMA_F32(acc2, a2, bf[kk]);
    WMMA_F32(acc3, a3, bf[kk]);
  }

  // C/D layout: lane<16 -> M=r, lane>=16 -> M=8+r; N = lane&15.
  #pragma unroll
  for (int r = 0; r < 8; ++r) {
    const int m = r + 8 * hi;
    const int gr = row0 + m;
    if (gr      < N) out[(size_t)(gr     ) * 128 + col0 + mn] = acc0[r];
    if (gr + 16 < N) out[(size_t)(gr + 16) * 128 + col0 + mn] = acc1[r];
    if (gr + 32 < N) out[(size_t)(gr + 32) * 128 + col0 + mn] = acc2[r];
    if (gr + 48 < N) out[(size_t)(gr + 48) * 128 + col0 + mn] = acc3[r];
  }
}

// ---------------- p = lin * dinv[row] (in place) ; agg init = p (self loop) ----------
__global__ __launch_bounds__(256) void k_scale2(float* __restrict__ buf,
                                                float* __restrict__ agg,
                                                const float* __restrict__ dinv, int total) {
  int i = blockIdx.x * 256 + threadIdx.x;
  if (i < total) {
    int v = i >> 7;
    float val = buf[i] * dinv[v];
    buf[i] = val;
    agg[i] = val;
  }
}

// ---------------- edge scatter: agg[dst] += p[src], one edge per wave32 -------------
// Each wave handles one edge (32 lanes x float4 = 128 features). src/dst are
// wave-uniform -> readfirstlane forces scalar (SMEM) edge reads: 32x less vector
// traffic on the dominant kernel.
__global__ __launch_bounds__(256) void k_scatter(const float4* __restrict__ p,
                                                 float* __restrict__ agg,
                                                 const long long* __restrict__ src,
                                                 const long long* __restrict__ dst, int E) {
  long long t = (long long)blockIdx.x * 256 + threadIdx.x;
  if (t >= (long long)E * 32) return;
  int e = __builtin_amdgcn_readfirstlane((int)(t >> 5));  // wave-uniform edge id
  int q = (int)(t & 31);                                  // float4 chunk 0..31
  int s = (int)src[e];
  int d = (int)dst[e];
  float4 v = p[(size_t)s * 32 + q];
  float* o = agg + (size_t)d * 128 + (size_t)q * 4;
  atomicAdd(o + 0, v.x);
  atomicAdd(o + 1, v.y);
  atomicAdd(o + 2, v.z);
  atomicAdd(o + 3, v.w);
}

// ---------------- layer-1 epilogue: h = relu(agg * dinv[row] + b1) (in place) -------
__global__ __launch_bounds__(256) void k_finish1(float* __restrict__ agg,
                                                 const float* __restrict__ dinv,
                                                 const float* __restrict__ b1, int total) {
  int i = blockIdx.x * 256 + threadIdx.x;
  if (i < total) {
    int v = i >> 7, j = i & 127;
    float val = agg[i] * dinv[v] + b1[j];
    agg[i] = fmaxf(val, 0.0f);
  }
}

// ---------------- layer-2 epilogue: split concatenated cols into mu / logvar --------
__global__ __launch_bounds__(256) void k_finish2(const float* __restrict__ agg,
                                                 const float* __restrict__ dinv,
                                                 const float* __restrict__ bmu,
                                                 const float* __restrict__ blv,
                                                 float* __restrict__ out, int N) {
  int i = blockIdx.x * 256 + threadIdx.x;
  int total = N * 128;
  if (i < total) {
    int v = i >> 7, j = i & 127;
    float val = agg[i] * dinv[v];
    if (j < 64) {
      out[(size_t)v * 64 + j] = val + bmu[j];                              // mu
    } else {
      out[(size_t)N * 64 + (size_t)v * 64 + (j - 64)] = val + blv[j - 64]; // logvar
    }
  }
}

// ---------------- launch ----------------

extern "C" void kernel_launch(void* const* d_in, const int* in_sizes, int n_in,
                              void* d_out, int out_size, void* d_ws, size_t ws_size,
                              hipStream_t stream) {
  const float* x       = (const float*)d_in[0];
  const long long* ei  = (const long long*)d_in[1];  // [2, E] int64
  const float* W1      = (const float*)d_in[2];
  const float* b1      = (const float*)d_in[3];
  const float* Wmu     = (const float*)d_in[4];
  const float* bmu     = (const float*)d_in[5];
  const float* Wlv     = (const float*)d_in[6];
  const float* blv     = (const float*)d_in[7];
  float* out = (float*)d_out;

  const int N = in_sizes[0] / 128;
  const int E = in_sizes[1] / 2;
  const long long* e_src = ei;
  const long long* e_dst = ei + E;

  // Workspace layout (all offsets 16B-aligned for N multiple of 4).
  char* w = (char*)d_ws;
  float* deg  = (float*)w; w += (size_t)N * 4;
  float* dinv = (float*)w; w += (size_t)N * 4;
  float* Wcat = (float*)w; w += (size_t)128 * 128 * 4;
  float* buf0 = (float*)w; w += (size_t)N * 128 * 4;  // lin / p
  float* buf1 = (float*)w;                            // agg / h

  const int T = 256;
  const int total = N * 128;
  const int gN   = (N + T - 1) / T;
  const int gE   = (E + T - 1) / T;
  const int gTot = (total + T - 1) / T;
  const int gTile = (N + 63) / 64;
  const long long scat_threads = (long long)E * 32;
  const int gScat = (int)((scat_threads + T - 1) / T);

  // Graph normalization (shared by both layers).
  k_fill  <<<gN, T, 0, stream>>>(deg, N, 1.0f);          // self loop
  k_degree<<<gE, T, 0, stream>>>(deg, e_dst, E);
  k_rsqrt <<<gN, T, 0, stream>>>(dinv, deg, N);
  k_wcat  <<<(128 * 128) / T, T, 0, stream>>>(Wcat, Wmu, Wlv);

  // Layer 1: h = relu(D^-1/2 (A+I) D^-1/2 (x@W1) + b1)
  k_gemm128<<<gTile, T, 0, stream>>>(x, W1, buf0, N);
  k_scale2 <<<gTot, T, 0, stream>>>(buf0, buf1, dinv, total);
  k_scatter<<<gScat, T, 0, stream>>>((const float4*)buf0, buf1, e_src, e_dst, E);
  k_finish1<<<gTot, T, 0, stream>>>(buf1, dinv, b1, total);

  // Layer 2 (mu and logvar fused via W_cat): single GEMM + single scatter pass.
  k_gemm128<<<gTile, T, 0, stream>>>(buf1, Wcat, buf0, N);
  k_scale2 <<<gTot, T, 0, stream>>>(buf0, buf1, dinv, total);
  k_scatter<<<gScat, T, 0, stream>>>((const float4*)buf0, buf1, e_src, e_dst, E);
  k_finish2<<<gTot, T, 0, stream>>>(buf1, dinv, bmu, blv, out, N);
}